// GatedShortBlock_85109071938303
// MI455X (gfx1250) — compile-verified
//
#include <hip/hip_runtime.h>

typedef __attribute__((ext_vector_type(2))) float v2f;
typedef __attribute__((ext_vector_type(8))) float v8f;

#define BB 256
#define DD 2048
#define KK 4

__device__ __forceinline__ v8f wmma_f32(v2f a, v2f b, v8f c) {
  // D = A(16x4, f32) * B(4x16, f32) + C(16x16, f32)
  return __builtin_amdgcn_wmma_f32_16x16x4_f32(false, a, false, b, (short)0, c,
                                               false, false);
}

// ---------------------------------------------------------------------------
// Kernel 1: fused gate GEMM (x @ w1^T, 3 gates) + B*x gate product +
// conv-cache roll/scatter + depthwise conv + C gate.
// Each wave: two 16-row (batch) tiles x one 16-col (feature) tile of all 3
// gates -> 6 accumulators, 5 loads : 6 WMMAs per k-step, software-pipelined
// one k-step ahead. Uniform base + 32-bit offsets => saddr-form loads.
// ---------------------------------------------------------------------------
__global__ __launch_bounds__(128) void gsb_gate_conv(
    const float* __restrict__ x, const float* __restrict__ conv_cache,
    const float* __restrict__ w1, const float* __restrict__ conv_w,
    const int* __restrict__ cache_pos, float* __restrict__ y_ws,
    float* __restrict__ state_out)
{
  const int lane = threadIdx.x & 31;
  const int wave = threadIdx.x >> 5;
  const int lh   = lane & 15;
  const int koff = (lane >> 4) << 1;              // lanes 16-31 hold K+2,K+3
  const int nb   = (blockIdx.x * 4 + wave) << 4;  // feature-tile base (n / d)
  const int mb   = blockIdx.y << 5;               // batch base: two 16-row tiles

  // Per-lane 32-bit element offsets against uniform bases (saddr addressing).
  const uint32_t a0o = (uint32_t)(mb + lh) * DD + koff;
  const uint32_t a1o = a0o + 16u * DD;
  const uint32_t g0o = (uint32_t)(nb + lh) * DD + koff;       // gate B rows
  const uint32_t g1o = g0o + (uint32_t)DD * DD;               // gate C rows
  const uint32_t g2o = g1o + (uint32_t)DD * DD;               // gate X rows

  // Prologue: fragments for k = 0.
  v2f a0 = *(const v2f*)(x + a0o);
  v2f a1 = *(const v2f*)(x + a1o);
  v2f b0 = *(const v2f*)(w1 + g0o);
  v2f b1 = *(const v2f*)(w1 + g1o);
  v2f b2 = *(const v2f*)(w1 + g2o);

  v8f acc[2][3] = {};  // [m-tile][gate: B, C, X]

  #pragma unroll 4
  for (int k = 4; k < DD; k += 4) {
    // Prefetch next k-step before consuming the current fragments.
    v2f na0 = *(const v2f*)(x + a0o + k);
    v2f na1 = *(const v2f*)(x + a1o + k);
    v2f nb0 = *(const v2f*)(w1 + g0o + k);
    v2f nb1 = *(const v2f*)(w1 + g1o + k);
    v2f nb2 = *(const v2f*)(w1 + g2o + k);
    acc[0][0] = wmma_f32(a0, b0, acc[0][0]);
    acc[0][1] = wmma_f32(a0, b1, acc[0][1]);
    acc[0][2] = wmma_f32(a0, b2, acc[0][2]);
    acc[1][0] = wmma_f32(a1, b0, acc[1][0]);
    acc[1][1] = wmma_f32(a1, b1, acc[1][1]);
    acc[1][2] = wmma_f32(a1, b2, acc[1][2]);
    a0 = na0; a1 = na1; b0 = nb0; b1 = nb1; b2 = nb2;
  }
  // Final k-step.
  acc[0][0] = wmma_f32(a0, b0, acc[0][0]);
  acc[0][1] = wmma_f32(a0, b1, acc[0][1]);
  acc[0][2] = wmma_f32(a0, b2, acc[0][2]);
  acc[1][0] = wmma_f32(a1, b0, acc[1][0]);
  acc[1][1] = wmma_f32(a1, b1, acc[1][1]);
  acc[1][2] = wmma_f32(a1, b2, acc[1][2]);

  int pos = cache_pos[0];
  pos = pos < 0 ? 0 : (pos > KK - 1 ? KK - 1 : pos);

  const int    d   = nb + lh;                     // feature index
  const float4 cw  = *(const float4*)(conv_w + (size_t)d * KK);
  const int    mhi = (lane >> 4) << 3;            // upper half-wave: M += 8

  #pragma unroll
  for (int t = 0; t < 2; ++t) {
    #pragma unroll
    for (int v = 0; v < 8; ++v) {
      const int    b  = mb + t * 16 + v + mhi;            // batch index
      const float  bx = acc[t][0][v] * acc[t][2][v];      // B-gate * x-gate
      const float4 c  = *(const float4*)(conv_cache + ((size_t)b * DD + d) * KK);
      // roll(cache, -1): state = [c1, c2, c3, c0]; then scatter bx at pos.
      float st0 = c.y, st1 = c.z, st2 = c.w, st3 = c.x;
      if      (pos == 0) st0 = bx;
      else if (pos == 1) st1 = bx;
      else if (pos == 2) st2 = bx;
      else               st3 = bx;
      const float conv_out = st0 * cw.x + st1 * cw.y + st2 * cw.z + st3 * cw.w;
      y_ws[(size_t)b * DD + d] = acc[t][1][v] * conv_out; // C gate
      float4 st; st.x = st0; st.y = st1; st.z = st2; st.w = st3;
      *(float4*)(state_out + ((size_t)b * DD + d) * KK) = st;
    }
  }
}

// ---------------------------------------------------------------------------
// Kernel 2: out = y @ w2^T (M=256, N=2048, K=2048).
// Each wave: 2x2 grid of 16x16 tiles -> 4 loads : 4 WMMAs per k-step,
// software-pipelined one k-step ahead.
// ---------------------------------------------------------------------------
__global__ __launch_bounds__(128) void gsb_out_proj(
    const float* __restrict__ y_ws, const float* __restrict__ w2,
    float* __restrict__ out)
{
  const int lane = threadIdx.x & 31;
  const int wave = threadIdx.x >> 5;
  const int lh   = lane & 15;
  const int koff = (lane >> 4) << 1;
  const int nb   = (blockIdx.x * 4 + wave) << 5;  // two adjacent n-tiles
  const int mb   = blockIdx.y << 5;               // two adjacent m-tiles

  const uint32_t a0o = (uint32_t)(mb + lh) * DD + koff;
  const uint32_t a1o = a0o + 16u * DD;
  const uint32_t w0o = (uint32_t)(nb + lh) * DD + koff;
  const uint32_t w1o = w0o + 16u * DD;

  v2f a0 = *(const v2f*)(y_ws + a0o);
  v2f a1 = *(const v2f*)(y_ws + a1o);
  v2f b0 = *(const v2f*)(w2 + w0o);
  v2f b1 = *(const v2f*)(w2 + w1o);

  v8f acc[2][2] = {};  // [m-tile][n-tile]

  #pragma unroll 4
  for (int k = 4; k < DD; k += 4) {
    v2f na0 = *(const v2f*)(y_ws + a0o + k);
    v2f na1 = *(const v2f*)(y_ws + a1o + k);
    v2f nb0 = *(const v2f*)(w2 + w0o + k);
    v2f nb1 = *(const v2f*)(w2 + w1o + k);
    acc[0][0] = wmma_f32(a0, b0, acc[0][0]);
    acc[0][1] = wmma_f32(a0, b1, acc[0][1]);
    acc[1][0] = wmma_f32(a1, b0, acc[1][0]);
    acc[1][1] = wmma_f32(a1, b1, acc[1][1]);
    a0 = na0; a1 = na1; b0 = nb0; b1 = nb1;
  }
  acc[0][0] = wmma_f32(a0, b0, acc[0][0]);
  acc[0][1] = wmma_f32(a0, b1, acc[0][1]);
  acc[1][0] = wmma_f32(a1, b0, acc[1][0]);
  acc[1][1] = wmma_f32(a1, b1, acc[1][1]);

  const int mhi = (lane >> 4) << 3;
  #pragma unroll
  for (int t = 0; t < 2; ++t) {
    #pragma unroll
    for (int v = 0; v < 8; ++v) {
      const int b = mb + t * 16 + v + mhi;
      out[(size_t)b * DD + nb + lh]      = acc[t][0][v];
      out[(size_t)b * DD + nb + 16 + lh] = acc[t][1][v];
    }
  }
}

// ---------------------------------------------------------------------------
// d_in order: x, conv_cache, w1, w2, conv_w, cache_pos_ids (all f32 / i32).
// d_out: out [B*D] floats, then state [B*D*K] floats.
// d_ws:  y [B*D] floats (2 MB).
// ---------------------------------------------------------------------------
extern "C" void kernel_launch(void* const* d_in, const int* in_sizes, int n_in,
                              void* d_out, int out_size, void* d_ws, size_t ws_size,
                              hipStream_t stream) {
  const float* x          = (const float*)d_in[0];
  const float* conv_cache = (const float*)d_in[1];
  const float* w1         = (const float*)d_in[2];
  const float* w2         = (const float*)d_in[3];
  const float* conv_w     = (const float*)d_in[4];
  const int*   pos        = (const int*)d_in[5];

  float* out   = (float*)d_out;               // [B, 1, D]
  float* state = out + (size_t)BB * DD;       // [B, D, K]
  float* y_ws  = (float*)d_ws;                // [B, D]

  // 128 n-tiles / 4 waves = 32 blocks x; 256 rows / 32 = 8 blocks y.
  dim3 g1(DD / 64, BB / 32);
  gsb_gate_conv<<<g1, 128, 0, stream>>>(x, conv_cache, w1, conv_w, pos, y_ws, state);

  // 2048 cols / (4 waves * 32) = 16 blocks x; 256 rows / 32 = 8 blocks y.
  dim3 g2(DD / 128, BB / 32);
  gsb_out_proj<<<g2, 128, 0, stream>>>(y_ws, w2, out);
}